// Graph_GCN_54829552500943
// MI455X (gfx1250) — compile-verified
//
#include <hip/hip_runtime.h>
#include <hip/hip_bf16.h>

// ---------------------------------------------------------------------------
// GCN forward pass for MI455X (gfx1250, wave32).
//   - Dense layers: fp32 WMMA (V_WMMA_F32_16X16X4_F32); weight-streaming bound
//     (~240 MB of weights -> ~10 us at 23.3 TB/s). Split-K for enough
//     concurrent HBM streams; deterministic workspace reduce (no fp atomics).
//   - Weight tiles double-buffered into LDS via the Tensor Data Mover
//     (in-order TENSORcnt -> s_wait_tensorcnt 1 overlaps DMA with WMMA).
//   - SpMM chain: L2-resident gather + global_atomic_add_f32 scatter.
// ---------------------------------------------------------------------------

#define GV   16384      // V: graph nodes
#define GB   64         // B: batch
#define GK   6          // Chebyshev order
#define GCF  32         // conv features
#define GFIN 65536      // FC1 fan-in = GCF * (GV/8)

#define TDM_KC   256    // K-chunk staged per TDM transfer (dwords per row)
#define TDM_PAD  2      // pad dwords appended per row by TDM
#define LDS_LDW  (TDM_KC + TDM_PAD)   // 258 floats: lane stride 2 banks
#define LDS_BUF  (16 * LDS_LDW)       // floats per buffer

typedef unsigned int u32;
typedef __attribute__((ext_vector_type(2))) float v2f;
typedef __attribute__((ext_vector_type(8))) float v8f;
typedef __attribute__((ext_vector_type(4))) u32   v4u;
typedef __attribute__((ext_vector_type(8))) int   v8i_;
typedef __attribute__((ext_vector_type(4))) int   v4i_;

// ----------------------------- utility kernels -----------------------------

__global__ void k_zero(float* p, int n) {
    int i = blockIdx.x * 256 + threadIdx.x;
    if (i < n) p[i] = 0.0f;
}

__global__ void k_negcopy(float* dst, const float* src, int n) {
    int i = blockIdx.x * 256 + threadIdx.x;
    if (i < n) dst[i] = -src[i];
}

// x_in [B, V] -> x0 [V, B]   (coalesced reads over v)
__global__ void k_transpose(const float* __restrict__ xin, float* __restrict__ x0) {
    unsigned idx = blockIdx.x * 256u + threadIdx.x;   // over V*B = 1M
    unsigned v = idx % GV;
    unsigned b = idx / GV;
    x0[(size_t)v * GB + b] = xin[(size_t)b * GV + v];
}

// out[rows[e], :] += scale * vals[e] * xin[cols[e], :]   (64 lanes = 64 cols)
__global__ void k_spmm(const float* __restrict__ vals,
                       const int*   __restrict__ rows,
                       const int*   __restrict__ cols,
                       const float* __restrict__ xin,
                       float* out, float scale, int E) {
    int e = blockIdx.x * 4 + (threadIdx.x >> 6);
    if (e >= E) return;
    int t = threadIdx.x & 63;
    float v = scale * vals[e];
    atomicAdd(out + (size_t)rows[e] * GB + t, v * xin[(size_t)cols[e] * GB + t]);
}

// Cheby feature combine + bias + (relu o maxpool-8).  relu(max) == max(relu).
__global__ void k_cheby_pool(const float* __restrict__ x0, const float* __restrict__ x1,
                             const float* __restrict__ x2, const float* __restrict__ x3,
                             const float* __restrict__ x4, const float* __restrict__ x5,
                             const float* __restrict__ cw,   // [32,6]
                             const float* __restrict__ cb,   // [32]
                             float* __restrict__ xp) {       // [B, 65536]
    const int vp = blockIdx.x;      // 0..2047
    const int b  = threadIdx.x;     // 0..63
    const float* xs[GK] = {x0, x1, x2, x3, x4, x5};
    float dmax[GCF];
    #pragma unroll
    for (int i = 0; i < 8; ++i) {
        const size_t voff = (size_t)(vp * 8 + i) * GB + b;
        float t[GK];
        #pragma unroll
        for (int k = 0; k < GK; ++k) t[k] = xs[k][voff];
        #pragma unroll
        for (int f = 0; f < GCF; ++f) {
            float d = cb[f];
            #pragma unroll
            for (int k = 0; k < GK; ++k) d = fmaf(t[k], cw[f * GK + k], d);
            dmax[f] = (i == 0) ? d : fmaxf(dmax[f], d);
        }
    }
    float* o = xp + (size_t)b * GFIN + (size_t)vp * GCF;
    #pragma unroll
    for (int f = 0; f < GCF; ++f) o[f] = fmaxf(dmax[f], 0.0f);
}

// ----------------------- TDM: stage W tile into LDS ------------------------
// Stages W[row0 .. row0+15][kc .. kc+TDM_KC) (f32, row stride K) into LDS with
// 2-dword row padding (rows land LDS_LDW floats apart -> ds_load_b64 from 16
// different rows hits 32 distinct banks).
// D# per CDNA5 ISA:
//   group0: [1:0]=count=1 | [63:32]=lds_addr | [120:64]=global tile addr |
//           [127:126]=type=2
//   group1: data_size=2(4B), pad_enable, pad_interval=7(256dw), pad_amount=1(2dw),
//           tensor_dim0=K, tensor_dim1=N, tile_dim0=TDM_KC, tile_dim1=16,
//           tensor_dim0_stride=K
__device__ __forceinline__ void tdm_load_wtile(const float* gsrc, u32 lds_byte,
                                               int K, int N) {
    unsigned long long ga = (unsigned long long)(uintptr_t)gsrc;
    v4u g0;
    g0[0] = 1u;                                   // count=1, user descriptor
    g0[1] = lds_byte;                             // LDS destination (bytes)
    g0[2] = (u32)ga;                              // global addr [31:0]
    g0[3] = (u32)((ga >> 32) & 0x01FFFFFFu) | (2u << 30);   // addr[56:32]|type=2
    v8i_ g1;
    g1[0] = (int)((2u << 16) | (1u << 20) | (7u << 22) | (1u << 25));
    g1[1] = (int)(((u32)K & 0xFFFFu) << 16);                  // tensor_dim0 lo
    g1[2] = (int)((((u32)K >> 16) & 0xFFFFu) |                // tensor_dim0 hi
                  (((u32)N & 0xFFFFu) << 16));                // tensor_dim1 lo
    g1[3] = (int)((((u32)N >> 16) & 0xFFFFu) |                // tensor_dim1 hi
                  ((u32)TDM_KC << 16));                       // tile_dim0
    g1[4] = 16;                                               // tile_dim1 (rows)
    g1[5] = (int)(u32)K;                                      // dim0_stride lo
    g1[6] = 0;                                                // stride hi, d1s lo
    g1[7] = 0;
    v4i_ z4 = {0, 0, 0, 0};
#if __clang_major__ >= 23
    v8i_ z8 = {0, 0, 0, 0, 0, 0, 0, 0};
    __builtin_amdgcn_tensor_load_to_lds(g0, g1, z4, z4, z8, 0);
#else
    __builtin_amdgcn_tensor_load_to_lds(g0, g1, z4, z4, 0);
#endif
}

// -------------------------- WMMA fp32 GEMM ---------------------------------
// Y[64, N] = act( X[64, K] @ W[N, K]^T + bias )
// Block = 128 threads = 4 waves; wave w owns rows m0=16w; blockIdx.x = n-tile;
// blockIdx.y = K-split (Kc elements each). The B tile (shared by the 4 waves)
// is double-buffered into LDS by the TDM: wave0 issues the next chunk's
// transfer before waiting (tensorcnt<=1 -> previous transfer landed, since
// per-wave tensor ops complete in order), overlapping DMA with WMMA. The
// end-of-chunk barrier guarantees no wave still reads a buffer when its next
// overwrite is issued. A streams from global (L2-resident activations).
// A lane layout (16x4 f32): lanes 0-15 -> K=k,k+1 ; lanes 16-31 -> K=k+2,k+3;
// B mirrors with the N index on lanes.
template <bool RELU, bool SPLIT>
__global__ void k_gemm_wmma(const float* __restrict__ X,
                            const float* __restrict__ W,
                            const float* __restrict__ bias,
                            float* __restrict__ Y,
                            int K, int N, int Kc) {
    __shared__ float lds_w[2 * LDS_BUF];

    const int lane = threadIdx.x & 31;
    const int wave = threadIdx.x >> 5;          // 0..3
    const int m0 = wave * 16;
    const int n0 = blockIdx.x * 16;
    const int lr = lane & 15;
    const int lg = lane >> 4;
    const int k_begin = blockIdx.y * Kc;
    const int k_end   = k_begin + Kc;           // Kc | K, TDM_KC | Kc

    const float* a_ptr = X + (size_t)(m0 + lr) * K + 2 * lg;
    const float* w_row = W + (size_t)n0 * K;
    const float* b_row = lds_w + lr * LDS_LDW + 2 * lg;
    const u32 lds_byte = (u32)(uintptr_t)lds_w;

    v8f acc = {0.f, 0.f, 0.f, 0.f, 0.f, 0.f, 0.f, 0.f};

    // prologue: transfer for the first chunk
    if (threadIdx.x < 32)
        tdm_load_wtile(w_row + k_begin, lds_byte, K, N);

    int cur = 0;
    for (int kc = k_begin; kc < k_end; kc += TDM_KC, cur ^= 1) {
        const bool has_next = (kc + TDM_KC < k_end);
        if (threadIdx.x < 32) {
            if (has_next) {
                // issue chunk i+1 into the other buffer, then wait for chunk i
                tdm_load_wtile(w_row + kc + TDM_KC,
                               lds_byte + (u32)((cur ^ 1) * LDS_BUF * 4), K, N);
                __builtin_amdgcn_s_wait_tensorcnt(1);
            } else {
                __builtin_amdgcn_s_wait_tensorcnt(0);
            }
        }
        __syncthreads();
        // prefetch the A stream one chunk ahead (global_prefetch_b8)
        if (has_next) __builtin_prefetch(a_ptr + kc + TDM_KC, 0, 1);

        const float* br = b_row + cur * LDS_BUF;
        #pragma unroll 8
        for (int k = 0; k < TDM_KC; k += 4) {
            v2f av = *(const v2f*)(a_ptr + kc + k);
            v2f bv = *(const v2f*)(br + k);
            // (neg_a, A, neg_b, B, c_mod, C, reuse_a, reuse_b)
            acc = __builtin_amdgcn_wmma_f32_16x16x4_f32(
                false, av, false, bv, (short)0, acc, false, false);
        }
        __syncthreads();    // all waves done with buf[cur] before it is reused
    }

    if (SPLIT) {
        float* P = Y + (size_t)blockIdx.y * ((size_t)GB * N);
        #pragma unroll
        for (int r = 0; r < 8; ++r)
            P[(size_t)(m0 + r + 8 * lg) * N + (n0 + lr)] = acc[r];
    } else {
        const float bn = bias[n0 + lr];
        #pragma unroll
        for (int r = 0; r < 8; ++r) {
            float v = acc[r] + bn;
            if (RELU) v = fmaxf(v, 0.0f);
            Y[(size_t)(m0 + r + 8 * lg) * N + (n0 + lr)] = v;
        }
    }
}

// Deterministic split-K reduce: Y[64,N] = act( sum_s P[s][64,N] + bias )
__global__ void k_reduce_bias_act(const float* __restrict__ P,
                                  const float* __restrict__ bias,
                                  float* __restrict__ Y,
                                  int S, int N, int relu) {
    int i = blockIdx.x * 256 + threadIdx.x;     // over 64*N
    if (i >= GB * N) return;
    float v = 0.0f;
    for (int s = 0; s < S; ++s) v += P[(size_t)s * GB * N + i];
    v += bias[i % N];
    if (relu) v = fmaxf(v, 0.0f);
    Y[i] = v;
}

// ------------------- classifier head: sum2 + log_softmax -------------------
__global__ void k_sum2_logsoftmax(const float* __restrict__ xh,   // [64,512]
                                  const float* __restrict__ xn,   // [64,512]
                                  const float* __restrict__ w,    // [10,1024]
                                  const float* __restrict__ bias, // [10]
                                  float* __restrict__ logp) {     // [64,10]
    const int b = blockIdx.x;
    const int lane = threadIdx.x;   // 0..31
    float lg[10];
    #pragma unroll
    for (int o = 0; o < 10; ++o) {
        float p = 0.0f;
        for (int k = lane; k < 1024; k += 32) {
            float xv = (k < 512) ? xh[b * 512 + k] : xn[b * 512 + (k - 512)];
            p = fmaf(xv, w[o * 1024 + k], p);
        }
        #pragma unroll
        for (int off = 16; off > 0; off >>= 1) p += __shfl_down(p, off);
        lg[o] = p + bias[o];
    }
    if (lane == 0) {
        float m = lg[0];
        #pragma unroll
        for (int o = 1; o < 10; ++o) m = fmaxf(m, lg[o]);
        float s = 0.0f;
        #pragma unroll
        for (int o = 0; o < 10; ++o) s += __expf(lg[o] - m);
        float lse = m + __logf(s);
        #pragma unroll
        for (int o = 0; o < 10; ++o) logp[b * 10 + o] = lg[o] - lse;
    }
}

// ------------------------------- launcher ----------------------------------

extern "C" void kernel_launch(void* const* d_in, const int* in_sizes, int n_in,
                              void* d_out, int out_size, void* d_ws, size_t ws_size,
                              hipStream_t stream) {
    const float* x_in   = (const float*)d_in[0];
    const float* L_vals = (const float*)d_in[1];
    const float* cl1_w  = (const float*)d_in[2];
    const float* cl1_b  = (const float*)d_in[3];
    const float* fc1_w  = (const float*)d_in[4];
    const float* fc1_b  = (const float*)d_in[5];
    const float* fc2_w  = (const float*)d_in[6];
    const float* fc2_b  = (const float*)d_in[7];
    const float* fc3_w  = (const float*)d_in[8];
    const float* fc3_b  = (const float*)d_in[9];
    const float* nn1_w  = (const float*)d_in[10];
    const float* nn1_b  = (const float*)d_in[11];
    const float* nn2_w  = (const float*)d_in[12];
    const float* nn2_b  = (const float*)d_in[13];
    const float* sum2_w = (const float*)d_in[14];
    const float* sum2_b = (const float*)d_in[15];
    const int*   L_rows = (const int*)d_in[16];
    const int*   L_cols = (const int*)d_in[17];
    const int E = in_sizes[1];

    // outputs: x_decode [64,16384] | x_hidden [64,512] | logp [64,10]
    float* out_f    = (float*)d_out;
    float* x_decode = out_f;
    float* x_hidden = out_f + (size_t)GB * GV;
    float* logp     = x_hidden + (size_t)GB * 512;

    // workspace layout (~47 MB)
    const size_t VB = (size_t)GV * GB;               // 1,048,576
    float* w  = (float*)d_ws;
    float* xs[GK];
    for (int k = 0; k < GK; ++k) xs[k] = w + k * VB; // 6 x 4 MB
    float* xp   = w + GK * VB;                       // [64, 65536]  16 MB
    float* xn1  = xp + (size_t)GB * GFIN;            // [64, 1024]
    float* xd   = xn1 + (size_t)GB * 1024;           // [64, 512]
    float* xn   = xd + (size_t)GB * 512;             // [64, 512]
    float* pfc1 = xn + (size_t)GB * 512;             // 16 x [64,512]  2 MB
    float* pnn1 = pfc1 + (size_t)16 * GB * 512;      //  8 x [64,1024] 2 MB

    const int nVB = (int)VB;

    // 1) x0 = x_in^T
    k_transpose<<<nVB / 256, 256, 0, stream>>>(x_in, xs[0]);

    // 2) x1 = L x0
    k_zero<<<nVB / 256, 256, 0, stream>>>(xs[1], nVB);
    k_spmm<<<(E + 3) / 4, 256, 0, stream>>>(L_vals, L_rows, L_cols, xs[0], xs[1], 1.0f, E);

    // 3) x_k = 2 L x_{k-1} - x_{k-2}
    for (int k = 2; k < GK; ++k) {
        k_negcopy<<<nVB / 256, 256, 0, stream>>>(xs[k], xs[k - 2], nVB);
        k_spmm<<<(E + 3) / 4, 256, 0, stream>>>(L_vals, L_rows, L_cols, xs[k - 1], xs[k], 2.0f, E);
    }

    // 4) combine K features, relu, pool-8  -> xp [64, 65536]
    k_cheby_pool<<<GV / 8, 64, 0, stream>>>(xs[0], xs[1], xs[2], xs[3], xs[4], xs[5],
                                            cl1_w, cl1_b, xp);

    // 5) dense stack (WMMA fp32 + TDM double-buffered weights)
    //    FC1: 134 MB weights -> split-K 16 (512 blocks)
    k_gemm_wmma<false, true><<<dim3(512 / 16, 16), 128, 0, stream>>>(
        xp, fc1_w, fc1_b, pfc1, GFIN, 512, GFIN / 16);
    k_reduce_bias_act<<<(GB * 512 + 255) / 256, 256, 0, stream>>>(
        pfc1, fc1_b, x_hidden, 16, 512, 1);
    //    FC2 direct
    k_gemm_wmma<true, false><<<dim3(512 / 16, 1), 128, 0, stream>>>(
        x_hidden, fc2_w, fc2_b, xd, 512, 512, 512);
    //    FC3: N=16384 -> 1024 blocks, no relu, straight to x_decode
    k_gemm_wmma<false, false><<<dim3(GV / 16, 1), 128, 0, stream>>>(
        xd, fc3_w, fc3_b, x_decode, 512, GV, 512);
    //    NN1: 67 MB weights -> split-K 8 (512 blocks)
    k_gemm_wmma<false, true><<<dim3(1024 / 16, 8), 128, 0, stream>>>(
        x_in, nn1_w, nn1_b, pnn1, GV, 1024, GV / 8);
    k_reduce_bias_act<<<(GB * 1024 + 255) / 256, 256, 0, stream>>>(
        pnn1, nn1_b, xn1, 8, 1024, 1);
    //    NN2 direct
    k_gemm_wmma<true, false><<<dim3(512 / 16, 1), 128, 0, stream>>>(
        xn1, nn2_w, nn2_b, xn, 1024, 512, 1024);

    // 6) fuse + classify + log_softmax
    k_sum2_logsoftmax<<<GB, 32, 0, stream>>>(x_hidden, xn, sum2_w, sum2_b, logp);
}